// SWD21_28449863369565
// MI455X (gfx1250) — compile-verified
//
#include <hip/hip_runtime.h>

// Butterfly-shift (per-column circular roll, shift_d = 0,0,2,4,...,252) followed by
// window-2 compare-exchange along L.  B=16, L=16384, D=128, fp32.
//
// Memory-bound kernel: stage a (R+64)-row x 32-col tile in LDS via CDNA5 async
// global->LDS b128 copies (ASYNCcnt), then do the per-column shifted gather out of
// LDS (bank-conflict-free) + min/max, with coalesced non-temporal b32 stores.

#define L_DIM 16384
#define D_DIM 128
#define B_DIM 16
#define R_TILE 256            // output rows per tile
#define HALO 64               // covers max in-group shift span (<=62) rounded up
#define LROWS (R_TILE + HALO) // 320 LDS rows
#define GCOLS 32              // columns per group (4 groups of 32)
#define NTHREADS 256          // 8 waves (wave32)

__global__ __launch_bounds__(NTHREADS)
void SWD21_butterfly_minmax(const float* __restrict__ v, float* __restrict__ out) {
  __shared__ float lds[LROWS * GCOLS]; // 40 KB

  const int r0 = blockIdx.x * R_TILE;  // first output row of tile
  const int gq = blockIdx.y;           // column group [0,4)
  const int b  = blockIdx.z;           // batch

  // max shift within this column group (shifts are even; d=0 has shift 0)
  const int maxshift = (gq == 0) ? 60 : (64 * gq + 60);
  const int ws = r0 - maxshift;        // first input row needed (may be negative -> mod L)

  const float* vb = v + (size_t)b * L_DIM * D_DIM + (size_t)gq * GCOLS;

  // ---------------- async load phase: LROWS rows x 32 cols = 2560 float4 ----------------
  // Each thread issues exactly 10 async b128 copies; consecutive lanes cover
  // consecutive 16B chunks (8 per 128B row segment) -> fully coalesced, each
  // HBM line touched once per tile.
#pragma unroll
  for (int it = 0; it < (LROWS * 8) / NTHREADS; ++it) {
    const int q     = it * NTHREADS + threadIdx.x;
    const int row_l = q >> 3;                       // LDS row
    const int c4    = q & 7;                        // float4 within row
    const int grow  = (ws + row_l) & (L_DIM - 1);   // circular row index
    const float* gp = vb + (size_t)grow * D_DIM + c4 * 4;
    const unsigned lds_addr =
        (unsigned)(unsigned long long)(const void*)&lds[row_l * GCOLS + c4 * 4];
    const unsigned long long ga = (unsigned long long)gp;
    // CDNA5 async direct-to-LDS 128-bit copy (tracked by ASYNCcnt, bypasses VGPRs)
    asm volatile("global_load_async_to_lds_b128 %0, %1, off"
                 :: "v"(lds_addr), "v"(ga)
                 : "memory");
  }
  // drain this wave's async copies, then sync the workgroup before consuming LDS
  asm volatile("s_wait_asynccnt 0" ::: "memory");
  __syncthreads();

  // ---------------- compute phase: shifted gather + compare-exchange ----------------
  const int d_local = threadIdx.x & (GCOLS - 1);
  const int d       = gq * GCOLS + d_local;
  const int shift   = (d == 0) ? 0 : 2 * (d - 1);   // <= 252, always even
  const int koff    = maxshift - shift;             // even, >= 0

  float* ob = out + (size_t)b * L_DIM * D_DIM + (size_t)gq * GCOLS + d_local;

  // 8 row-pairs per iteration (one pair per wave), 16 iterations.
  // LDS banks = (k*32 + d_local) % 64 with k even across all lanes of a wave
  // -> banks reduce to (d_local + const) % 64: conflict-free.
#pragma unroll
  for (int it = 0; it < (R_TILE / 2) / (NTHREADS / 32); ++it) {
    const int p  = it * (NTHREADS / 32) + (int)(threadIdx.x >> 5);
    const int i0 = 2 * p;          // even output row within tile
    const int k  = i0 + koff;      // LDS row of the gathered even input row
    const float a = lds[k * GCOLS + d_local];
    const float c = lds[(k + 1) * GCOLS + d_local];
    const float mn = fminf(a, c);
    const float mx = fmaxf(a, c);
    const size_t o = (size_t)(r0 + i0) * D_DIM;
    // Output is write-once: non-temporal stores keep the write stream from
    // evicting input halo lines out of the 192MB L2 (halo reuse across tiles).
    __builtin_nontemporal_store(mn, &ob[o]);          // out[b, r0+i0,   d]
    __builtin_nontemporal_store(mx, &ob[o + D_DIM]);  // out[b, r0+i0+1, d]
  }
}

extern "C" void kernel_launch(void* const* d_in, const int* in_sizes, int n_in,
                              void* d_out, int out_size, void* d_ws, size_t ws_size,
                              hipStream_t stream) {
  (void)in_sizes; (void)n_in; (void)d_ws; (void)ws_size; (void)out_size;
  const float* v = (const float*)d_in[0];
  float* out = (float*)d_out;
  dim3 grid(L_DIM / R_TILE, D_DIM / GCOLS, B_DIM); // 64 x 4 x 16 = 4096 blocks
  dim3 block(NTHREADS);
  SWD21_butterfly_minmax<<<grid, block, 0, stream>>>(v, out);
}